// EGATLayer_79714593014207
// MI455X (gfx1250) — compile-verified
//
#include <hip/hip_runtime.h>
#include <math.h>

#define Bn 8
#define Nn 2048
#define Mn 3072
#define FEn 64
#define Dn 128
#define NEGV (-1.0e9f)

#define EC_CHUNKS 48
#define EC_ROWS   (Mn / EC_CHUNKS)   // 64

typedef __attribute__((ext_vector_type(2))) float v2f;
typedef __attribute__((ext_vector_type(8))) float v8f;

// ---------------- block reductions (256 threads) ----------------
__device__ __forceinline__ float block_reduce_sum(float v, float* red, int t) {
    red[t] = v; __syncthreads();
    for (int o = 128; o > 0; o >>= 1) {
        if (t < o) red[t] = red[t] + red[t + o];
        __syncthreads();
    }
    float r = red[0]; __syncthreads();
    return r;
}
__device__ __forceinline__ float block_reduce_max(float v, float* red, int t) {
    red[t] = v; __syncthreads();
    for (int o = 128; o > 0; o >>= 1) {
        if (t < o) red[t] = fmaxf(red[t], red[t + o]);
        __syncthreads();
    }
    float r = red[0]; __syncthreads();
    return r;
}

// ---------------- fp32 WMMA GEMM: C[b] (Mrows x Dn) = A[b] (Mrows x K) @ Bm[b] (K x Dn) ----
// One wave computes a 16-row x 128-col stripe: 8 v8f accumulators, 8 WMMAs per K-step,
// A fragment loaded once per K-step (b64 load), reused across all 8 column tiles.
__global__ __launch_bounds__(128) void wmma_gemm_f32_kernel(
    const float* __restrict__ A, long long sAb, int lda, int K,
    const float* __restrict__ Bm, long long sBb,
    float* __restrict__ C, long long sCb)
{
    const int ldb = Dn, ldc = Dn;
    const int wave = threadIdx.x >> 5;
    const int lane = threadIdx.x & 31;
    const int tile = blockIdx.x * 4 + wave;     // 16-row tile index
    const int b    = blockIdx.y;
    const int half = lane >> 4;                 // K sub-pair selector
    const int l16  = lane & 15;

    const float* Arow = A + (long long)b * sAb + (long long)(tile * 16 + l16) * lda;
    const v2f*   Arow2 = (const v2f*)Arow;      // 8B-aligned: lda even, k multiple of 4
    const float* Bb   = Bm + (long long)b * sBb;

    v8f acc[8];
#pragma unroll
    for (int t = 0; t < 8; ++t) acc[t] = (v8f)0.0f;

    for (int k = 0; k < K; k += 4) {
        // A 16x4 f32 fragment: lane(l16,half) -> row=l16, K = k + 2*half + {0,1}
        v2f a = Arow2[(k >> 1) + half];
        // B 4x16 fragments: lane(l16,half) -> col=l16, K rows k+2*half and k+2*half+1
        const float* B0 = Bb + (long long)(k + 2 * half) * ldb + l16;
#pragma unroll
        for (int t = 0; t < 8; ++t) {
            v2f bv;
            bv.x = B0[t * 16];
            bv.y = B0[t * 16 + ldb];
            acc[t] = __builtin_amdgcn_wmma_f32_16x16x4_f32(
                false, a, false, bv, (short)0, acc[t], false, false);
        }
    }

    // C/D 16x16 layout: VGPR r holds row (r + 8*half), col = l16
    float* Crow = C + (long long)b * sCb + (long long)(tile * 16) * ldc;
#pragma unroll
    for (int t = 0; t < 8; ++t)
#pragma unroll
        for (int r = 0; r < 8; ++r)
            Crow[(long long)(r + 8 * half) * ldc + t * 16 + l16] = acc[t][r];
}

// ---------------- ec stage 1: partial column sums over 64-row chunks ----------------
__global__ __launch_bounds__(Dn) void ec_part_kernel(const float* __restrict__ Et,
                                                     float* __restrict__ part)
{
    int b = blockIdx.x, ch = blockIdx.y, d = threadIdx.x;
    const float* p = Et + (long long)b * Mn * Dn + (long long)ch * EC_ROWS * Dn + d;
    float s = 0.f;
#pragma unroll 4
    for (int m = 0; m < EC_ROWS; ++m) s += p[(long long)m * Dn];
    part[((long long)b * EC_CHUNKS + ch) * Dn + d] = s;
}

// ---------------- ec stage 2: reduce partials -> mean ----------------
__global__ __launch_bounds__(Dn) void ec_final_kernel(const float* __restrict__ part,
                                                      float* __restrict__ ec)
{
    int b = blockIdx.x, d = threadIdx.x;
    const float* p = part + (long long)b * EC_CHUNKS * Dn + d;
    float s = 0.f;
#pragma unroll
    for (int c = 0; c < EC_CHUNKS; ++c) s += p[c * Dn];
    ec[b * Dn + d] = s * (1.0f / (float)Mn);
}

// ---------------- scalars: s1, s2, c[b], r[b]=b3*h_avg[b] ----------------
__global__ __launch_bounds__(256) void scal_kernel(
    const float* __restrict__ WH, const float* __restrict__ av,
    const float* __restrict__ bvec, const float* __restrict__ bet,
    const float* __restrict__ ec, float* __restrict__ scal)
{
    __shared__ float red[256];
    int t = threadIdx.x;
    if (blockIdx.x == 0) {
        float v1 = (t < Dn) ? WH[t] * av[t]      : 0.f;
        float v2 = (t < Dn) ? WH[t] * av[Dn + t] : 0.f;
        float s1 = block_reduce_sum(v1, red, t);
        float s2 = block_reduce_sum(v2, red, t);
        if (t == 0) { scal[0] = s1; scal[1] = s2; }
    } else {
        int b = blockIdx.x - 1;
        float vc = (t < Dn) ? ec[b * Dn + t] * av[2 * Dn + t] : 0.f;
        float cb = block_reduce_sum(vc, red, t);
        float hs = 0.f;
        for (int n = t; n < Nn; n += 256) hs += bet[b * Nn + n];
        float havg = block_reduce_sum(hs, red, t) * (1.0f / (float)Nn);
        if (t == 0) { scal[2 + b] = cb; scal[10 + b] = bvec[2 * Dn] * havg; }
    }
}

// ---------------- p[row]=E_t[row]·b1, q[row]=E_t[row]·b2 (one wave per row) ----------------
__global__ __launch_bounds__(256) void pq_kernel(const float* __restrict__ Et,
                                                 const float* __restrict__ bvec,
                                                 float* __restrict__ p,
                                                 float* __restrict__ q)
{
    int lane = threadIdx.x & 31;
    int row  = blockIdx.x * 8 + (threadIdx.x >> 5);   // global row in [0, B*M)
    const float* e = Et + (long long)row * Dn + lane * 4;
    float sp = 0.f, sq = 0.f;
#pragma unroll
    for (int i = 0; i < 4; ++i) {
        float x = e[i];
        sp += x * bvec[lane * 4 + i];
        sq += x * bvec[Dn + lane * 4 + i];
    }
#pragma unroll
    for (int o = 16; o > 0; o >>= 1) {
        sp += __shfl_xor(sp, o, 32);
        sq += __shfl_xor(sq, o, 32);
    }
    if (lane == 0) { p[row] = sp; q[row] = sq; }
}

// ---------------- node attention row: alpha + H_new + Hm (rank-1 collapse) ----------------
__global__ __launch_bounds__(256) void node_attn_kernel(
    const float* __restrict__ bet, const int* __restrict__ adj,
    const float* __restrict__ scal, const float* __restrict__ WH,
    const float* __restrict__ ec, float* __restrict__ alpha,
    float* __restrict__ Hnew, float* __restrict__ Hm)
{
    __shared__ float red[256];
    int i = blockIdx.x, b = blockIdx.y, t = threadIdx.x;
    const float* bb = bet + b * Nn;
    float ui = bb[i] * scal[0] + scal[2 + b];
    float s2 = scal[1];
    const int* arow = adj + ((long long)b * Nn + i) * (long long)Nn;

    float x[8], bj[8];
    float mx = NEGV;
#pragma unroll
    for (int kk = 0; kk < 8; ++kk) {
        int j = t + kk * 256;
        float bv = bb[j]; bj[kk] = bv;
        float v = (arow[j] > 0) ? tanhf(ui + bv * s2) : NEGV;
        x[kk] = v; mx = fmaxf(mx, v);
    }
    mx = block_reduce_max(mx, red, t);
    float s = 0.f, sb = 0.f;
#pragma unroll
    for (int kk = 0; kk < 8; ++kk) {
        float e = expf(x[kk] - mx);
        x[kk] = e; s += e; sb += e * bj[kk];
    }
    s  = block_reduce_sum(s,  red, t);
    sb = block_reduce_sum(sb, red, t);
    float inv = 1.0f / s;
    float* orow = alpha + ((long long)b * Nn + i) * (long long)Nn;
#pragma unroll
    for (int kk = 0; kk < 8; ++kk) orow[t + kk * 256] = x[kk] * inv;

    if (t < Dn) {
        float w  = sb * inv;                 // sum_j alpha*bet
        float hv = w * WH[t];                // H_new row = w * WH
        long long o = ((long long)b * Nn + i) * Dn + t;
        Hnew[o] = hv;
        Hm[o]   = ec[b * Dn] * hv;           // ec[b,0] * H_new
    }
}

// ---------------- edge attention row: beta ----------------
__global__ __launch_bounds__(256) void edge_attn_kernel(
    const int* __restrict__ adj, const float* __restrict__ p,
    const float* __restrict__ q, const float* __restrict__ scal,
    float* __restrict__ beta)
{
    __shared__ float red[256];
    int i = blockIdx.x, b = blockIdx.y, t = threadIdx.x;
    float gi = p[b * Mn + i] + scal[10 + b];
    const float* qb = q + b * Mn;
    const int* arow = adj + ((long long)b * Mn + i) * (long long)Mn;

    float x[12];
    float mx = NEGV;
#pragma unroll
    for (int kk = 0; kk < 12; ++kk) {
        int j = t + kk * 256;
        float v = (arow[j] > 0) ? tanhf(gi + qb[j]) : NEGV;
        x[kk] = v; mx = fmaxf(mx, v);
    }
    mx = block_reduce_max(mx, red, t);
    float s = 0.f;
#pragma unroll
    for (int kk = 0; kk < 12; ++kk) {
        float e = expf(x[kk] - mx);
        x[kk] = e; s += e;
    }
    s = block_reduce_sum(s, red, t);
    float inv = 1.0f / s;
    float* orow = beta + ((long long)b * Mn + i) * (long long)Mn;
#pragma unroll
    for (int kk = 0; kk < 12; ++kk) orow[t + kk * 256] = x[kk] * inv;
}

// ---------------- launcher ----------------
extern "C" void kernel_launch(void* const* d_in, const int* in_sizes, int n_in,
                              void* d_out, int out_size, void* d_ws, size_t ws_size,
                              hipStream_t stream)
{
    const float* bet  = (const float*)d_in[0];   // (B,N)
    const float* LF   = (const float*)d_in[1];   // (B,M,FE)
    const int*   nadj = (const int*)  d_in[2];   // (B,N,N)
    const int*   eadj = (const int*)  d_in[3];   // (B,M,M)
    const float* WH   = (const float*)d_in[4];   // (1,D)
    const float* WE   = (const float*)d_in[5];   // (FE,D)
    const float* av   = (const float*)d_in[6];   // (3D,1)
    const float* bvec = (const float*)d_in[7];   // (2D+1,1)

    // outputs, concatenated: H_new, E_new, Hm, alpha, beta
    float* out   = (float*)d_out;
    float* Hnew  = out;
    float* Enew  = Hnew + (size_t)Bn * Nn * Dn;
    float* Hm    = Enew + (size_t)Bn * Mn * Dn;
    float* alpha = Hm   + (size_t)Bn * Nn * Dn;
    float* beta  = alpha + (size_t)Bn * Nn * Nn;

    // workspace
    float* ws   = (float*)d_ws;
    float* Et   = ws;                               // B*M*D
    float* ec   = Et + (size_t)Bn * Mn * Dn;        // B*D
    float* pv   = ec + (size_t)Bn * Dn;             // B*M
    float* qv   = pv + (size_t)Bn * Mn;             // B*M
    float* scal = qv + (size_t)Bn * Mn;             // 32
    float* ecp  = scal + 32;                        // B*EC_CHUNKS*D partials

    // K1: E_t = LF @ WE  (WMMA fp32)
    dim3 g1(Mn / 64, Bn);
    wmma_gemm_f32_kernel<<<g1, 128, 0, stream>>>(
        LF, (long long)Mn * FEn, FEn, FEn, WE, 0LL, Et, (long long)Mn * Dn);

    // K2: ec = mean_m E_t  (two-stage deterministic reduction)
    dim3 g2(Bn, EC_CHUNKS);
    ec_part_kernel<<<g2, Dn, 0, stream>>>(Et, ecp);
    ec_final_kernel<<<Bn, Dn, 0, stream>>>(ecp, ec);

    // K3: scalars
    scal_kernel<<<Bn + 1, 256, 0, stream>>>(WH, av, bvec, bet, ec, scal);

    // K4: p, q row dots
    pq_kernel<<<(Bn * Mn) / 8, 256, 0, stream>>>(Et, bvec, pv, qv);

    // K5: node softmax rows -> alpha, H_new, Hm
    dim3 g5(Nn, Bn);
    node_attn_kernel<<<g5, 256, 0, stream>>>(bet, nadj, scal, WH, ec, alpha, Hnew, Hm);

    // K6: edge softmax rows -> beta
    dim3 g6(Mn, Bn);
    edge_attn_kernel<<<g6, 256, 0, stream>>>(eadj, pv, qv, scal, beta);

    // K7: E_new = beta @ E_t  (WMMA fp32, K = M)
    wmma_gemm_f32_kernel<<<g1, 128, 0, stream>>>(
        beta, (long long)Mn * Mn, Mn, Mn, Et, (long long)Mn * Dn, Enew, (long long)Mn * Dn);
}